// UniBIP_33517924778601
// MI455X (gfx1250) — compile-verified
//
#include <hip/hip_runtime.h>

#define N_NODES 100000
#define N_EDGES 640000
#define D_X 128
#define D_E 32
#define D_OUT 128

typedef __attribute__((ext_vector_type(2))) float v2f;
typedef __attribute__((ext_vector_type(8))) float v8f;

// TDM descriptor SGPR-group types (probe-verified builtin signatures)
typedef __attribute__((ext_vector_type(4))) unsigned int tdm_u4;
typedef __attribute__((ext_vector_type(8))) int tdm_i8;
typedef __attribute__((ext_vector_type(4))) int tdm_i4;

// D = A(16x4) * B(4x16) + C(16x16), all f32, wave32.
__device__ __forceinline__ v8f wmma_f32_4(v2f a, v2f b, v8f c) {
    return __builtin_amdgcn_wmma_f32_16x16x4_f32(
        /*neg_a=*/false, a, /*neg_b=*/false, b,
        /*c_mod=*/(short)0, c, /*reuse_a=*/false, /*reuse_b=*/false);
}

// ---------------------------------------------------------------------------
// Tensor Data Mover: DMA a 2-D f32 tile (tile_h rows x tile_w elems, row
// stride row_stride_elems) from global memory into LDS. Builds the D# per
// CDNA5 ISA ch.8: group0 = {count, lds_addr, global_addr, type=2},
// group1 = {data_size=4B, tensor dims, tile dims, dim0 stride},
// groups 2/3 zero (2-D: tile_dim2/3 = 0 = unused). Tracked by TENSORcnt.
// ---------------------------------------------------------------------------
__device__ __forceinline__ void tdm_load_tile_f32(void* lds_dst,
                                                  const void* gsrc,
                                                  unsigned tile_w,
                                                  unsigned tile_h,
                                                  unsigned row_stride_elems) {
    const unsigned long long ga = (unsigned long long)(uintptr_t)gsrc;
    tdm_u4 g0;
    g0.x = 1u;                                    // count=1, user descriptor
    g0.y = (unsigned)(uintptr_t)lds_dst;          // LDS byte address (low 32b)
    g0.z = (unsigned)ga;                          // global_addr[31:0]
    g0.w = ((unsigned)(ga >> 32) & 0x01FFFFFFu)   // global_addr[56:32]
           | (2u << 30);                          // type=2 ("image")
    tdm_i8 g1;
    g1[0] = (int)(2u << 16);                      // data_size=4B; no multicast
    g1[1] = (int)((row_stride_elems & 0xFFFFu) << 16);        // tensor_dim0 lo
    g1[2] = (int)((row_stride_elems >> 16) |
                  ((tile_h & 0xFFFFu) << 16));    // tensor_dim0 hi | dim1 lo
    g1[3] = (int)((tile_h >> 16) |
                  ((tile_w & 0xFFFFu) << 16));    // tensor_dim1 hi | tile_dim0
    g1[4] = (int)(tile_h & 0xFFFFu);              // tile_dim1; tile_dim2=0
    g1[5] = (int)row_stride_elems;                // tensor_dim0_stride[31:0]
    g1[6] = 0;                                    // stride0 hi; stride1 lo
    g1[7] = 0;                                    // stride1 hi
    tdm_i4 z4 = {0, 0, 0, 0};
#if defined(__clang_major__) && __clang_major__ >= 23
    tdm_i8 z8 = {0, 0, 0, 0, 0, 0, 0, 0};
    __builtin_amdgcn_tensor_load_to_lds(g0, g1, z4, z4, z8, 0);
#else
    __builtin_amdgcn_tensor_load_to_lds(g0, g1, z4, z4, 0);
#endif
}

// ---------------------------------------------------------------------------
// Kernel 1: fuse weights (tiny, plain FMA).
//   Wc1[128x128] = Wx @ Wm_top ; Wc2[32x128] = We @ Wm_bot
//   bc[128] = bx@Wm_top + be@Wm_bot + bm
// ---------------------------------------------------------------------------
__global__ void __launch_bounds__(128)
fuse_weights(const float* __restrict__ Wx, const float* __restrict__ bx,
             const float* __restrict__ We, const float* __restrict__ be,
             const float* __restrict__ Wm, const float* __restrict__ bm,
             float* __restrict__ Wc1, float* __restrict__ Wc2,
             float* __restrict__ bc) {
    const int j = threadIdx.x;
    const int blk = blockIdx.x;
    if (blk < D_X) {
        float acc = 0.f;
        for (int k = 0; k < D_OUT; ++k)
            acc = fmaf(Wx[blk * D_OUT + k], Wm[k * D_OUT + j], acc);
        Wc1[blk * D_OUT + j] = acc;
    } else if (blk < D_X + D_E) {
        const int i = blk - D_X;
        float acc = 0.f;
        for (int k = 0; k < D_OUT; ++k)
            acc = fmaf(We[i * D_OUT + k], Wm[(D_OUT + k) * D_OUT + j], acc);
        Wc2[i * D_OUT + j] = acc;
    } else {
        float acc = bm[j];
        for (int k = 0; k < D_OUT; ++k) {
            acc = fmaf(bx[k], Wm[k * D_OUT + j], acc);
            acc = fmaf(be[k], Wm[(D_OUT + k) * D_OUT + j], acc);
        }
        bc[j] = acc;
    }
}

// ---------------------------------------------------------------------------
// Kernel 2: t1 = x @ Wc1   [100000 x 128], WMMA f32 16x16x4.
// Block = 8 waves; block -> 16-node tile, wave -> 16-col tile.
// x tile staged into LDS by the Tensor Data Mover.
// ---------------------------------------------------------------------------
__global__ void __launch_bounds__(256)
node_gemm(const float* __restrict__ x, const float* __restrict__ Wc1,
          float* __restrict__ t1) {
    __shared__ float sX[16 * D_X];                // 8 KB
    const int tid = threadIdx.x;
    const int node0 = blockIdx.x * 16;

    if (tid == 0)                                 // wave 0 kicks the DMA
        tdm_load_tile_f32(sX, x + (size_t)node0 * D_X,
                          /*tile_w=*/D_X, /*tile_h=*/16,
                          /*row_stride=*/D_X);
    if (tid < 32)                                 // issuing wave drains it
        __builtin_amdgcn_s_wait_tensorcnt(0);
    __syncthreads();

    const int wave = tid >> 5;
    const int lane = tid & 31;
    const int half = lane >> 4;
    const int l    = lane & 15;
    const int n    = wave * 16 + l;

    v8f c = {};
    for (int k = 0; k < D_X; k += 4) {
        const int ka = k + 2 * half;
        v2f a, b;
        a.x = sX[l * D_X + ka];                   // A[m=l][ka]
        a.y = sX[l * D_X + ka + 1];
        b.x = Wc1[ka * D_OUT + n];                // B[ka][n]  (L2-resident)
        b.y = Wc1[(ka + 1) * D_OUT + n];
        c = wmma_f32_4(a, b, c);
    }
    float* dst = t1 + (size_t)node0 * D_OUT + n;
#pragma unroll
    for (int j = 0; j < 8; ++j)
        dst[(size_t)(j + 8 * half) * D_OUT] = c[j];
}

// ---------------------------------------------------------------------------
// Kernel 3: per-edge  msg = leaky_relu(t1[src] + ea@Wc2 + bc); atomic
// scatter into agg[dst]. Wc2 staged by TDM; t1 rows gathered coalesced.
// ---------------------------------------------------------------------------
__global__ void __launch_bounds__(256)
edge_msg(const float* __restrict__ ea, const int* __restrict__ eidx,
         const float* __restrict__ t1, const float* __restrict__ Wc2,
         const float* __restrict__ bc, float* __restrict__ agg) {
    __shared__ float sEA[16 * D_E];               // 2 KB
    __shared__ float sW[D_E * D_OUT];             // 16 KB (TDM-filled)
    __shared__ float sT1[16 * D_OUT];             // 8 KB
    __shared__ float sBC[D_OUT];
    __shared__ int sSrc[16];
    __shared__ int sDst[16];

    const int tid = threadIdx.x;
    const int E0 = blockIdx.x * 16;

    if (tid == 0)                                 // DMA Wc2 -> LDS
        tdm_load_tile_f32(sW, Wc2, /*tile_w=*/D_OUT, /*tile_h=*/D_E,
                          /*row_stride=*/D_OUT);

    if (tid < 16)       sSrc[tid] = eidx[E0 + tid];
    else if (tid < 32)  sDst[tid - 16] = eidx[N_EDGES + E0 + (tid - 16)];
    if (tid < D_OUT)    sBC[tid] = bc[tid];
    if (tid < 16 * D_E / 4)
        ((float4*)sEA)[tid] = ((const float4*)(ea + (size_t)E0 * D_E))[tid];
    __syncthreads();

    // coalesced gather of the 16 t1 rows (needs sSrc)
    for (int i = tid; i < 16 * D_OUT / 4; i += 256) {
        const int row = i >> 5, c4 = i & 31;
        ((float4*)sT1)[i] =
            ((const float4*)(t1 + (size_t)sSrc[row] * D_OUT))[c4];
    }
    if (tid < 32)                                 // wave 0 drains the DMA
        __builtin_amdgcn_s_wait_tensorcnt(0);
    __syncthreads();

    const int wave = tid >> 5;
    const int lane = tid & 31;
    const int half = lane >> 4;
    const int l    = lane & 15;
    const int n    = wave * 16 + l;

    v8f c = {};
#pragma unroll
    for (int k = 0; k < D_E; k += 4) {
        const int ka = k + 2 * half;
        v2f a, b;
        a.x = sEA[l * D_E + ka];
        a.y = sEA[l * D_E + ka + 1];
        b.x = sW[ka * D_OUT + n];
        b.y = sW[(ka + 1) * D_OUT + n];
        c = wmma_f32_4(a, b, c);
    }

#pragma unroll
    for (int j = 0; j < 8; ++j) {
        const int m = j + 8 * half;
        float v = c[j] + sT1[m * D_OUT + n] + sBC[n];
        float r = v > 0.f ? v : 0.01f * v;        // leaky_relu
        atomicAdd(agg + (size_t)sDst[m] * D_OUT + n, r);
    }
}

// ---------------------------------------------------------------------------
// Kernel 0/4: zero accumulator; final out = sigmoid(agg) * relu(beta)
// ---------------------------------------------------------------------------
__global__ void __launch_bounds__(256)
zero_out(float4* __restrict__ p, int n4) {
    const int i = blockIdx.x * blockDim.x + threadIdx.x;
    if (i < n4) p[i] = make_float4(0.f, 0.f, 0.f, 0.f);
}

__global__ void __launch_bounds__(256)
finalize(float* __restrict__ agg, const float* __restrict__ beta, int n) {
    const int i = blockIdx.x * blockDim.x + threadIdx.x;
    if (i < n) {
        const float s = fmaxf(beta[0], 0.f);
        const float v = agg[i];
        agg[i] = s / (1.f + __expf(-v));
    }
}

// ---------------------------------------------------------------------------
extern "C" void kernel_launch(void* const* d_in, const int* in_sizes, int n_in,
                              void* d_out, int out_size, void* d_ws,
                              size_t ws_size, hipStream_t stream) {
    (void)in_sizes; (void)n_in; (void)out_size; (void)ws_size;
    const float* x    = (const float*)d_in[0];
    const float* ea   = (const float*)d_in[1];
    const float* Wx   = (const float*)d_in[2];
    const float* bx   = (const float*)d_in[3];
    const float* We   = (const float*)d_in[4];
    const float* be   = (const float*)d_in[5];
    const float* Wm   = (const float*)d_in[6];
    const float* bm   = (const float*)d_in[7];
    const float* beta = (const float*)d_in[8];
    const int*   eidx = (const int*)d_in[9];

    // workspace layout (floats): Wc1 | Wc2 | bc | t1   (~51.3 MB total)
    float* ws  = (float*)d_ws;
    float* Wc1 = ws;                               // 128*128
    float* Wc2 = ws + D_X * D_OUT;                 // 32*128
    float* bc  = Wc2 + D_E * D_OUT;                // 128
    float* t1  = bc + D_OUT;                       // N_NODES*128 (16B aligned)

    float* agg = (float*)d_out;                    // accumulator == output

    fuse_weights<<<D_X + D_E + 1, 128, 0, stream>>>(Wx, bx, We, be, Wm, bm,
                                                    Wc1, Wc2, bc);
    const int n_out = N_NODES * D_OUT;
    zero_out<<<(n_out / 4 + 255) / 256, 256, 0, stream>>>((float4*)agg,
                                                          n_out / 4);
    node_gemm<<<N_NODES / 16, 256, 0, stream>>>(x, Wc1, t1);
    edge_msg<<<N_EDGES / 16, 256, 0, stream>>>(ea, eidx, t1, Wc2, bc, agg);
    finalize<<<(n_out + 255) / 256, 256, 0, stream>>>(agg, beta, n_out);
}